// RNN_11072425689786
// MI455X (gfx1250) — compile-verified
//
#include <hip/hip_runtime.h>
#include <hip/hip_bf16.h>

#define DIMS    1024
#define NSTEPS  16384
#define TILE    16

// ---- scan configuration ----
#define SCAN_WGS      32                 // persistent workgroups (one per WGP)
#define ROWS_PER_WG   (DIMS / SCAN_WGS)  // 32 rows of W_hh per WG
#define SCAN_THREADS  512                // 16 waves of 32

typedef __attribute__((ext_vector_type(2))) float        v2f;
typedef __attribute__((ext_vector_type(8))) float        v8f;
typedef __attribute__((ext_vector_type(4))) unsigned int v4u;
typedef __attribute__((ext_vector_type(4))) int          v4i;
typedef __attribute__((ext_vector_type(8))) int          v8i;

// ---------------------------------------------------------------------------
// Kernel A: xi = X @ W_hi^T + b, written into d_out (consumed in place later).
// One wave computes one 16x16 output tile via V_WMMA_F32_16X16X4_F32,
// accumulating K=1024 in chunks of 4 (exact f32 arithmetic).
//
// ISA fragment layouts (CDNA5 ISA):
//   A (16x4):  lane L, vgpr v -> row = L%16, col = (L/16)*2 + v   (v2f load)
//   B (4x16):  lane L, vgpr v -> col = L%16, row = (L/16)*2 + v
//   C/D:       lane L, vgpr r -> row = r + (L/16)*8, col = L%16
// B[k][n] = W_hi^T[k][n] = W_hi[n][k]  (so B frag is also a contiguous pair).
// ---------------------------------------------------------------------------
__global__ __launch_bounds__(256) void rnn_xproj_wmma(
    const float* __restrict__ X, const float* __restrict__ Whi,
    const float* __restrict__ bias, float* __restrict__ xi)
{
    const int lane = threadIdx.x & 31;
    const int wave = threadIdx.x >> 5;          // 0..7
    const int m0   = blockIdx.x * TILE;         // timestep-tile
    const int n0   = (blockIdx.y * 8 + wave) * TILE;
    const int grp  = lane >> 4;                 // 0 or 1 (half-wave)
    const int lan  = lane & 15;

    const float* arow = X   + (size_t)(m0 + lan) * DIMS + grp * 2;
    const float* brow = Whi + (size_t)(n0 + lan) * DIMS + grp * 2;

    v8f acc = {};
#pragma unroll 4
    for (int k = 0; k < DIMS; k += 4) {
        v2f a = *(const v2f*)(arow + k);
        v2f b = *(const v2f*)(brow + k);
        // 8 args: (neg_a, A, neg_b, B, c_mod, C, reuse_a, reuse_b)
        acc = __builtin_amdgcn_wmma_f32_16x16x4_f32(
                  false, a, false, b, (short)0, acc, false, false);
    }

    const float bn = bias[n0 + lan];
    float* orow = xi + (size_t)(m0 + grp * 8) * DIMS + n0 + lan;
#pragma unroll
    for (int r = 0; r < 8; ++r)
        orow[(size_t)r * DIMS] = acc[r] + bn;
}

// ---------------------------------------------------------------------------
// Init: h double-buffer gets h0; barrier counter reset (deterministic launch).
// ---------------------------------------------------------------------------
__global__ void rnn_scan_init(const float* __restrict__ h0,
                              float* __restrict__ hbuf,
                              unsigned* __restrict__ cnt)
{
    int i = threadIdx.x;
    if (i < DIMS) hbuf[i] = h0[i];
    if (i == 0) *cnt = 0u;
}

// ---------------------------------------------------------------------------
// Kernel B: persistent sequential scan.
//   - Each WG pins its 32 rows of W_hh in LDS (128 KB), staged by the Tensor
//     Data Mover (TENSOR_LOAD_TO_LDS) as one 2D tile descriptor.
//   - Per step: 4 KB h broadcast via GLOBAL_LOAD_ASYNC_TO_LDS_B64 (ASYNCcnt),
//     32 wave-level dot products, tanh, residual output, one device barrier.
//   - `out` holds xi on entry; each WG reads its 32 xi values for step t and
//     overwrites those exact locations with the final output (safe in-place).
// ---------------------------------------------------------------------------
__global__ __launch_bounds__(SCAN_THREADS) void rnn_scan(
    const float* __restrict__ X, const float* __restrict__ Whh,
    float* __restrict__ out,            // xi in, final output out
    float* __restrict__ hbuf,           // 2*DIMS double buffer
    unsigned* __restrict__ cnt)         // barrier counter
{
    __shared__ float w_sh[ROWS_PER_WG * DIMS];  // 128 KB, resident all scan
    __shared__ float h_sh[DIMS];                // 4 KB

    const int wg   = blockIdx.x;
    const int tid  = threadIdx.x;
    const int lane = tid & 31;
    const int wave = tid >> 5;                  // 0..15
    const int row0 = wg * ROWS_PER_WG;

    // ---- Stage this WG's W_hh rows into LDS once via the Tensor Data Mover.
    // D# per CDNA5 ISA: 2D tensor, f32 elements,
    // tile = ROWS_PER_WG x DIMS starting at global_addr (tile start).
    if (wave == 0) {                           // one wave issues the DMA
        const unsigned long long gaddr =
            (unsigned long long)(uintptr_t)(Whh + (size_t)row0 * DIMS);
        const unsigned lds_base = (unsigned)(uintptr_t)w_sh;

        v4u g0;
        g0.x = 0x1u;                                     // count=1, user D#
        g0.y = lds_base;                                 // lds_addr (bytes)
        g0.z = (unsigned)(gaddr & 0xFFFFFFFFu);          // global_addr[31:0]
        g0.w = (unsigned)((gaddr >> 32) & 0x01FFFFFFu)   // global_addr[56:32]
               | (2u << 30);                             // type = 2 ("image")

        v8i g1;
        g1[0] = (int)(2u << 16);          // wg_mask=0, data_size=2 (4 bytes)
        g1[1] = (int)((unsigned)DIMS << 16);          // tensor_dim0[15:0]
        g1[2] = (int)((unsigned)DIMS << 16);          // dim0[31:16]=0 | tensor_dim1[15:0]
        g1[3] = (int)((unsigned)DIMS << 16);          // dim1[31:16]=0 | tile_dim0=1024
        g1[4] = ROWS_PER_WG;              // tile_dim1=32 | tile_dim2=0
        g1[5] = DIMS;                     // tensor_dim0_stride[31:0]
        g1[6] = 0;                        // stride0[47:32] | dim1_stride[15:0]
        g1[7] = 0;

        v4i gz4  = {0, 0, 0, 0};                   // groups 2/3 unused (2D)
        v8i gz8  = {0, 0, 0, 0, 0, 0, 0, 0};       // trailing arg (clang-23 form)

        // amdgpu-toolchain (clang-23) 6-arg form:
        // (v4u g0, v8i g1, v4i g2, v4i g3, v8i extra, i32 cpol)
        __builtin_amdgcn_tensor_load_to_lds(g0, g1, gz4, gz4, gz8, 0);
        __builtin_amdgcn_s_wait_tensorcnt(0);
    }
    __syncthreads();

    int p = 0;
    for (int t = 0; t < NSTEPS; ++t) {
        // Broadcast current h (4 KB, L2-resident) straight into LDS with the
        // async global->LDS path (no VGPR round-trip, tracked by ASYNCcnt).
        {
            const unsigned lds_off = (unsigned)(uintptr_t)&h_sh[2 * tid];
            const unsigned long long src =
                (unsigned long long)(uintptr_t)&hbuf[p * DIMS + 2 * tid];
            asm volatile("global_load_async_to_lds_b64 %0, %1, off"
                         :: "v"(lds_off), "v"(src) : "memory");
        }
        // Hint: pull next step's xi chunk toward the caches early.
        if (tid == 0 && t + 1 < NSTEPS)
            __builtin_prefetch(&out[(size_t)(t + 1) * DIMS + row0], 0, 1);
        asm volatile("s_wait_asynccnt 0x0" ::: "memory");
        __syncthreads();

        // 16 waves x 2 rows each = 32 rows of W_hh @ h.
#pragma unroll
        for (int rr = 0; rr < 2; ++rr) {
            const int r = wave * 2 + rr;
            const float* wrow = &w_sh[r * DIMS];
            float sum = 0.f;
#pragma unroll 8
            for (int k = lane; k < DIMS; k += 32)
                sum = fmaf(wrow[k], h_sh[k], sum);
            // wave32 butterfly reduction
            for (int off = 16; off > 0; off >>= 1)
                sum += __shfl_xor(sum, off, 32);
            if (lane == 0) {
                const int g = row0 + r;
                const size_t idx = (size_t)t * DIMS + g;
                const float y = tanhf(out[idx] + sum);   // out[idx] == xi_t[g]
                hbuf[(1 - p) * DIMS + g] = y;            // next h
                out[idx] = y + X[idx];                   // residual output
            }
        }

        // ---- device-wide barrier (one per step) ----
        __threadfence();
        __syncthreads();
        // Cluster fast-path if dispatched as a workgroup cluster; per ISA this
        // is an S_NOP when ClusterID==0, so it is safe unconditionally.
        __builtin_amdgcn_s_cluster_barrier();
        if (tid == 0) {
            __hip_atomic_fetch_add(cnt, 1u, __ATOMIC_ACQ_REL,
                                   __HIP_MEMORY_SCOPE_AGENT);
            const unsigned want = (unsigned)(t + 1) * SCAN_WGS;
            while (__hip_atomic_load(cnt, __ATOMIC_ACQUIRE,
                                     __HIP_MEMORY_SCOPE_AGENT) < want) {
                __builtin_amdgcn_s_sleep(8);
            }
        }
        __syncthreads();
        p ^= 1;
    }
}

// ---------------------------------------------------------------------------
extern "C" void kernel_launch(void* const* d_in, const int* in_sizes, int n_in,
                              void* d_out, int out_size, void* d_ws, size_t ws_size,
                              hipStream_t stream)
{
    const float* X   = (const float*)d_in[0];   // [N, DIMS]
    const float* h0  = (const float*)d_in[1];   // [DIMS]
    const float* Whi = (const float*)d_in[2];   // [DIMS, DIMS]
    const float* Whh = (const float*)d_in[3];   // [DIMS, DIMS]
    const float* b   = (const float*)d_in[4];   // [DIMS]

    float*    out  = (float*)d_out;
    float*    hbuf = (float*)d_ws;                                   // 2*DIMS f32
    unsigned* cnt  = (unsigned*)((char*)d_ws + 2 * DIMS * sizeof(float));

    // Phase 1: xi = X @ W_hi^T + b  -> d_out  (WMMA f32)
    dim3 gA(NSTEPS / TILE, (DIMS / TILE) / 8);   // (1024, 8), 8 waves/block
    rnn_xproj_wmma<<<gA, 256, 0, stream>>>(X, Whi, b, out);

    // Phase 2 init: h <- h0, barrier counter <- 0
    rnn_scan_init<<<1, 1024, 0, stream>>>(h0, hbuf, cnt);

    // Phase 3: persistent sequential scan, weights staged by TDM, LDS-resident
    rnn_scan<<<SCAN_WGS, SCAN_THREADS, 0, stream>>>(X, Whh, out, hbuf, cnt);
}